// GaussianMixtureHead_83683142795325
// MI455X (gfx1250) — compile-verified
//
#include <hip/hip_runtime.h>
#include <hip/hip_bf16.h>

// GaussianMixtureHead on MI455X (gfx1250, wave32).
//
// maha[n,c] = ||W_c x_n - v_c||^2 with W_c = L_c^{-1}, v_c = W_c mu_c.
// Heavy phase: 100 GEMMs (32768 x 128 x 128) on V_WMMA_F32_16X16X4_F32
// (fp32 matrix path preserves fp32 precision).  ~107 GFLOP, ~37 MB traffic
// (L2-resident) -> compute-limited.
//
// Round-3 change: hoist the 8 B fragments of each K-step into a register
// array before the WMMA burst so the ds_load_2addr_b64s issue as one clause
// and drain with staged s_wait_dscnt (overlap LDS latency with WMMA).

#define DIMD 128
#define NUMC 100
#define NSMP 32768

typedef __attribute__((ext_vector_type(2))) float v2f;
typedef __attribute__((ext_vector_type(8))) float v8f;

// ---------------------------------------------------------------------------
// Phase 1: per-class prep.  One block per class, 128 threads.
// Thread t forward-substitutes column t of W = L^{-1} (columns independent).
// Outputs:
//   WtP[c]  : W^T pair-interleaved: WtP[(k>>1)*256 + d*2 + (k&1)] = W[d][k]
//   Vv[c][d]: (W mu)[d]
//   Cst[c]  : log_pi[c] - D/2*log(2pi) - sum(log diag L)
// ---------------------------------------------------------------------------
__global__ __launch_bounds__(128) void gmh_prep(
    const float* __restrict__ means, const float* __restrict__ scale_tril,
    const float* __restrict__ log_pi, float* __restrict__ WtP,
    float* __restrict__ Vv, float* __restrict__ Cst) {
  const int c = blockIdx.x;
  const int t = threadIdx.x;

  __shared__ __align__(16) float sL[DIMD * DIMD];        // 64 KB
  __shared__ __align__(16) float sW[DIMD * (DIMD + 1)];  // 64.5 KB (padded)
  __shared__ float sRed[DIMD];

  const float* L = scale_tril + (size_t)c * DIMD * DIMD;
  for (int i = t; i < DIMD * DIMD; i += DIMD) sL[i] = L[i];  // coalesced
  __syncthreads();

  // Forward substitution: solve L * w = e_t  -> column t of W.
  for (int i = 0; i < t; ++i) sW[i * (DIMD + 1) + t] = 0.0f;
  sW[t * (DIMD + 1) + t] = 1.0f / sL[t * DIMD + t];
  for (int i = t + 1; i < DIMD; ++i) {
    float s = 0.0f;
    for (int k = t; k < i; ++k) s += sL[i * DIMD + k] * sW[k * (DIMD + 1) + t];
    sW[i * (DIMD + 1) + t] = -s / sL[i * DIMD + i];
  }
  __syncthreads();

  // Pair-interleaved W^T: idx -> kpair = idx>>8, d = (idx&255)>>1,
  // parity = idx&1; value = W[d][kpair*2+parity].
  float* wt = WtP + (size_t)c * DIMD * DIMD;
  for (int idx = t; idx < DIMD * DIMD; idx += DIMD) {
    const int kpair = idx >> 8;
    const int rem = idx & 255;
    const int d = rem >> 1;
    const int parity = rem & 1;
    wt[idx] = sW[d * (DIMD + 1) + kpair * 2 + parity];
  }

  // v[t] = sum_k W[t][k] * mu[k]  (W lower triangular -> k <= t)
  const float* mu = means + (size_t)c * DIMD;
  float v = 0.0f;
  for (int k = 0; k <= t; ++k) v += sW[t * (DIMD + 1) + k] * mu[k];
  Vv[(size_t)c * DIMD + t] = v;

  // Per-class constant.
  sRed[t] = logf(sL[t * DIMD + t]);
  __syncthreads();
  for (int off = DIMD / 2; off > 0; off >>= 1) {
    if (t < off) sRed[t] += sRed[t + off];
    __syncthreads();
  }
  if (t == 0) {
    const float half_d_log2pi = 0.5f * 128.0f * 1.8378770664093453f;
    Cst[c] = log_pi[c] - half_d_log2pi - sRed[0];
  }
}

// ---------------------------------------------------------------------------
// Phase 2: main GEMM + epilogue.
// grid = (NSMP/256, NUMC); block = 256 threads = 8 waves; wave owns 32 rows.
// A fragment (ISA 32-bit A 16x4): lanes 0-15 -> M=lane, K={k0,k0+1};
// lanes 16-31 -> M=lane-16, K={k0+2,k0+3}.  One global b64 per lane per tile.
// B fragments: 4x ds_load_2addr_b64 per K-step, hoisted ahead of 16 WMMAs.
// ---------------------------------------------------------------------------
__global__ __launch_bounds__(256) void gmh_main(
    const float* __restrict__ X, const float* __restrict__ WtP,
    const float* __restrict__ Vv, const float* __restrict__ Cst,
    float* __restrict__ out) {
  const int c = blockIdx.y;
  const int tid = threadIdx.x;

  __shared__ __align__(16) float sW[DIMD * DIMD];  // pair-interleaved W^T
  __shared__ float sV[DIMD];

  {  // cooperative stage (float4, coalesced)
    const float4* src = (const float4*)(WtP + (size_t)c * DIMD * DIMD);
    float4* dst = (float4*)sW;
    for (int i = tid; i < DIMD * DIMD / 4; i += 256) dst[i] = src[i];
    if (tid < DIMD) sV[tid] = Vv[(size_t)c * DIMD + tid];
  }
  __syncthreads();

  const int wave = tid >> 5;
  const int lane = tid & 31;
  const int half = lane >> 4;  // K sub-pair select
  const int mcol = lane & 15;  // A row in tile / output column in tile
  const int row0 = blockIdx.x * 256 + wave * 32;

  const float* xrow0 = X + (size_t)(row0 + mcol) * DIMD + half * 2;
  const float* xrow1 = xrow0 + (size_t)16 * DIMD;

  v8f acc[2][8] = {};  // 2 M-tiles x 8 D-tiles of 16x16 fp32

#pragma unroll 2
  for (int k0 = 0; k0 < DIMD; k0 += 4) {
    const v2f a0 = *(const v2f*)(xrow0 + k0);
    const v2f a1 = *(const v2f*)(xrow1 + k0);
    // pairIdx = (k0 + half*2)>>1 = k0/2 + half
    const float* wb = sW + ((k0 >> 1) + half) * 256 + mcol * 2;
    v2f b[8];
#pragma unroll
    for (int t = 0; t < 8; ++t) b[t] = *(const v2f*)(wb + t * 32);
#pragma unroll
    for (int t = 0; t < 8; ++t) {
      acc[0][t] = __builtin_amdgcn_wmma_f32_16x16x4_f32(
          false, a0, false, b[t], (short)0, acc[0][t], false, false);
      acc[1][t] = __builtin_amdgcn_wmma_f32_16x16x4_f32(
          false, a1, false, b[t], (short)0, acc[1][t], false, false);
    }
  }

  const float cst = Cst[c];

  // Epilogue.  C/D layout: VGPR r -> (M = r + half*8, N = mcol).
#pragma unroll
  for (int g = 0; g < 2; ++g) {
    const int rbase = row0 + g * 16 + half * 8;
#pragma unroll
    for (int r = 0; r < 8; ++r) {
      float s = 0.0f;
#pragma unroll
      for (int t = 0; t < 8; ++t) {
        const float y = acc[g][t][r] - sV[t * 16 + mcol];
        s += y * y;
      }
      // reduce over 16 lanes (masks < 16 stay within each half-wave)
      for (int m = 1; m < 16; m <<= 1) s += __shfl_xor(s, m, 32);
      if (mcol == 0) {
        out[(size_t)(rbase + r) * NUMC + c] = cst - 0.5f * s;
      }
    }
  }
}

extern "C" void kernel_launch(void* const* d_in, const int* in_sizes, int n_in,
                              void* d_out, int out_size, void* d_ws,
                              size_t ws_size, hipStream_t stream) {
  const float* features   = (const float*)d_in[0];  // (32768,128)
  const float* means      = (const float*)d_in[1];  // (100,128)
  const float* scale_tril = (const float*)d_in[2];  // (100,128,128)
  const float* log_pi     = (const float*)d_in[3];  // (100,)
  float* out = (float*)d_out;                       // (32768,100)

  float* WtP = (float*)d_ws;                        // C*D*D floats (6.55 MB)
  float* Vv  = WtP + (size_t)NUMC * DIMD * DIMD;    // C*D floats
  float* Cst = Vv + (size_t)NUMC * DIMD;            // C floats

  gmh_prep<<<dim3(NUMC), dim3(128), 0, stream>>>(means, scale_tril, log_pi,
                                                 WtP, Vv, Cst);
  gmh_main<<<dim3(NSMP / 256, NUMC), dim3(256), 0, stream>>>(features, WtP,
                                                             Vv, Cst, out);
}